// GCNEncoder_5738076307739
// MI455X (gfx1250) — compile-verified
//
#include <hip/hip_runtime.h>

typedef float v2f __attribute__((ext_vector_type(2)));
typedef float v8f __attribute__((ext_vector_type(8)));

#define IN_DIM 768
#define HID    256
#define OUT_DIM 384
#define LN_EPS 1e-5f

// ---------------------------------------------------------------------------
// Tiled fp32 WMMA GEMM:  D[M,Nc] = epilogue(A[M,K] @ W[K,Nc])
// 256 threads = 8 wave32 waves. Block tile 64x64, wave tile 16x32
// (two V_WMMA_F32_16X16X4_F32 tiles per wave). K staged in 16-chunks via LDS.
// flags: bit0 = add bias, bit1 = relu
// ---------------------------------------------------------------------------
__global__ __launch_bounds__(256) void gemm_wmma_f32(
    const float* __restrict__ A, int lda,
    const float* __restrict__ W, int ldw,
    const float* __restrict__ bias,
    float* __restrict__ D, int ldd,
    int M, int K, int flags)
{
    __shared__ float As[64][17];   // pad 17: conflict-free column reads
    __shared__ float Bs[16][72];   // pad 72: half-waves hit disjoint banks

    const int tid     = threadIdx.x;
    const int rowBase = blockIdx.x * 64;
    const int colBase = blockIdx.y * 64;

    const int wave = tid >> 5;
    const int lane = tid & 31;
    const int wr   = wave >> 1;     // 0..3 : 16-row band
    const int wc   = wave & 1;      // 0..1 : 32-col band
    const int r    = lane & 15;
    const int hi   = lane >> 4;     // half-wave select

    v8f acc0 = {};
    v8f acc1 = {};

    // staging decomposition (float4 per thread)
    const int ar = tid >> 2;               // 0..63
    const int ac = (tid & 3) * 4;          // 0,4,8,12
    const int br = tid >> 4;               // 0..15
    const int bc = (tid & 15) * 4;         // 0..60

    for (int k0 = 0; k0 < K; k0 += 16) {
        // A tile 64x16 (row-guarded, zero-fill)
        {
            int grow = rowBase + ar;
            float4 v = make_float4(0.f, 0.f, 0.f, 0.f);
            if (grow < M)
                v = *(const float4*)(A + (size_t)grow * lda + k0 + ac);
            As[ar][ac + 0] = v.x; As[ar][ac + 1] = v.y;
            As[ar][ac + 2] = v.z; As[ar][ac + 3] = v.w;
        }
        // B tile 16x64 (K,Nc always in range)
        {
            float4 v = *(const float4*)(W + (size_t)(k0 + br) * ldw + colBase + bc);
            Bs[br][bc + 0] = v.x; Bs[br][bc + 1] = v.y;
            Bs[br][bc + 2] = v.z; Bs[br][bc + 3] = v.w;
        }
        __syncthreads();

        #pragma unroll
        for (int kk = 0; kk < 16; kk += 4) {
            const int kA0 = kk + (hi ? 2 : 0);
            const int kA1 = kk + (hi ? 3 : 1);
            v2f a, b0, b1;
            a.x  = As[wr * 16 + r][kA0];
            a.y  = As[wr * 16 + r][kA1];
            b0.x = Bs[kA0][wc * 32 + r];
            b0.y = Bs[kA1][wc * 32 + r];
            b1.x = Bs[kA0][wc * 32 + 16 + r];
            b1.y = Bs[kA1][wc * 32 + 16 + r];
            acc0 = __builtin_amdgcn_wmma_f32_16x16x4_f32(
                       false, a, false, b0, (short)0, acc0, false, false);
            acc1 = __builtin_amdgcn_wmma_f32_16x16x4_f32(
                       false, a, false, b1, (short)0, acc1, false, false);
        }
        __syncthreads();
    }

    // epilogue: D row = rowBase + wr*16 + 8*hi + v ; col = colBase + wc*32 + r (+16)
    const int c0 = colBase + wc * 32 + r;
    const int c1 = c0 + 16;
    const float bv0 = (flags & 1) ? bias[c0] : 0.f;
    const float bv1 = (flags & 1) ? bias[c1] : 0.f;
    #pragma unroll
    for (int v = 0; v < 8; ++v) {
        int row = rowBase + wr * 16 + hi * 8 + v;
        if (row < M) {
            float o0 = acc0[v] + bv0;
            float o1 = acc1[v] + bv1;
            if (flags & 2) { o0 = fmaxf(o0, 0.f); o1 = fmaxf(o1, 0.f); }
            D[(size_t)row * ldd + c0] = o0;
            D[(size_t)row * ldd + c1] = o1;
        }
    }
}

// ---------------------------------------------------------------------------
// Degree / dinv kernels:  dinv[n] = rsqrt(1 + indegree(n))
// ---------------------------------------------------------------------------
__global__ void deg_init(float* deg, int n) {
    int i = blockIdx.x * blockDim.x + threadIdx.x;
    if (i < n) deg[i] = 1.0f;                 // self loop
}
__global__ void deg_count(const int* __restrict__ dst, float* deg, int e) {
    int i = blockIdx.x * blockDim.x + threadIdx.x;
    if (i < e) atomicAdd(&deg[dst[i]], 1.0f);
}
__global__ void deg_finish(float* deg, int n) {   // in place -> dinv
    int i = blockIdx.x * blockDim.x + threadIdx.x;
    if (i < n) deg[i] = rsqrtf(deg[i]);
}

// ---------------------------------------------------------------------------
// agg[n,:] = t[n,:] * dinv[n]^2 + bias   (self-loop term + bias init)
// one thread per 4 channels (float4)
// ---------------------------------------------------------------------------
__global__ __launch_bounds__(256) void agg_init(
    const float* __restrict__ t, const float* __restrict__ dinv,
    const float* __restrict__ bias, float* __restrict__ agg, int n)
{
    int i = blockIdx.x * blockDim.x + threadIdx.x;   // over n * (HID/4)
    int node = i >> 6;
    int c    = (i & 63) * 4;
    if (node >= n) return;
    float di = dinv[node];
    float cf = di * di;
    float4 v = *(const float4*)(t + (size_t)node * HID + c);
    float4 o;
    o.x = v.x * cf + bias[c + 0];
    o.y = v.y * cf + bias[c + 1];
    o.z = v.z * cf + bias[c + 2];
    o.w = v.w * cf + bias[c + 3];
    *(float4*)(agg + (size_t)node * HID + c) = o;
}

// ---------------------------------------------------------------------------
// Edge scatter: one wave32 per edge; gather 256-wide row of t[src], scale by
// dinv[src]*dinv[dst], atomic-add into agg[dst].
// ---------------------------------------------------------------------------
__global__ __launch_bounds__(256) void edge_scatter(
    const int* __restrict__ src, const int* __restrict__ dst,
    const float* __restrict__ dinv, const float* __restrict__ t,
    float* __restrict__ agg, int e)
{
    int gw   = (blockIdx.x * blockDim.x + threadIdx.x) >> 5;
    int lane = threadIdx.x & 31;
    if (gw >= e) return;
    int s = src[gw];
    int d = dst[gw];
    float coef = dinv[s] * dinv[d];
    const float* ts = t   + (size_t)s * HID;
    float*       ad = agg + (size_t)d * HID;
    #pragma unroll
    for (int it = 0; it < 2; ++it) {
        int c = it * 128 + lane * 4;
        float4 v = *(const float4*)(ts + c);
        atomicAdd(ad + c + 0, v.x * coef);
        atomicAdd(ad + c + 1, v.y * coef);
        atomicAdd(ad + c + 2, v.z * coef);
        atomicAdd(ad + c + 3, v.w * coef);
    }
}

// ---------------------------------------------------------------------------
// In-place LayerNorm + ReLU over 256-wide rows; one wave32 per row,
// 8 channels per lane, wave shuffle reduction.
// ---------------------------------------------------------------------------
__global__ __launch_bounds__(256) void ln_relu(
    float* __restrict__ h, const float* __restrict__ g,
    const float* __restrict__ be, int n)
{
    int row  = (blockIdx.x * blockDim.x + threadIdx.x) >> 5;
    int lane = threadIdx.x & 31;
    if (row >= n) return;
    float* p = h + (size_t)row * HID;
    int c = lane * 8;
    float4 v0 = *(const float4*)(p + c);
    float4 v1 = *(const float4*)(p + c + 4);
    float vv[8] = {v0.x, v0.y, v0.z, v0.w, v1.x, v1.y, v1.z, v1.w};

    float s = 0.f, sq = 0.f;
    #pragma unroll
    for (int j = 0; j < 8; ++j) { s += vv[j]; sq += vv[j] * vv[j]; }
    #pragma unroll
    for (int m = 16; m > 0; m >>= 1) {
        s  += __shfl_xor(s,  m, 32);
        sq += __shfl_xor(sq, m, 32);
    }
    float mean = s * (1.0f / HID);
    float var  = sq * (1.0f / HID) - mean * mean;
    float inv  = rsqrtf(var + LN_EPS);

    float o[8];
    #pragma unroll
    for (int j = 0; j < 8; ++j) {
        float y = (vv[j] - mean) * inv * g[c + j] + be[c + j];
        o[j] = fmaxf(y, 0.f);
    }
    *(float4*)(p + c)     = make_float4(o[0], o[1], o[2], o[3]);
    *(float4*)(p + c + 4) = make_float4(o[4], o[5], o[6], o[7]);
}

// ---------------------------------------------------------------------------
extern "C" void kernel_launch(void* const* d_in, const int* in_sizes, int n_in,
                              void* d_out, int out_size, void* d_ws, size_t ws_size,
                              hipStream_t stream)
{
    const float* x     = (const float*)d_in[0];
    const int*   ei    = (const int*)  d_in[1];
    const float* W_in  = (const float*)d_in[2];
    const float* b_in  = (const float*)d_in[3];
    const float* W1    = (const float*)d_in[4];
    const float* b1    = (const float*)d_in[5];
    const float* g1    = (const float*)d_in[6];
    const float* be1   = (const float*)d_in[7];
    const float* W2    = (const float*)d_in[8];
    const float* b2    = (const float*)d_in[9];
    const float* g2    = (const float*)d_in[10];
    const float* be2   = (const float*)d_in[11];
    const float* W_out = (const float*)d_in[12];
    const float* b_out = (const float*)d_in[13];

    const int N = in_sizes[0] / IN_DIM;
    const int E = in_sizes[1] / 2;
    const int* src = ei;
    const int* dst = ei + E;

    // workspace: dinv[N] | buf0[N*HID] (h / agg) | buf1[N*HID] (t = h @ W)
    float* dinv = (float*)d_ws;
    float* buf0 = dinv + ((size_t)N + 64);
    float* buf1 = buf0 + (size_t)N * HID;

    const int T = 256;
    dim3 blk(T);
    const int gN   = (N + T - 1) / T;
    const int gE   = (E + T - 1) / T;
    const int gRow = (N + 63) / 64;                   // GEMM row blocks
    const int gAgg = ((size_t)N * (HID / 4) + T - 1) / T;
    const int gWaveN = (N + 7) / 8;                   // wave-per-row kernels
    const int gWaveE = (E + 7) / 8;                   // wave-per-edge

    // dinv = rsqrt(1 + indegree)
    deg_init  <<<gN, blk, 0, stream>>>(dinv, N);
    deg_count <<<gE, blk, 0, stream>>>(dst, dinv, E);
    deg_finish<<<gN, blk, 0, stream>>>(dinv, N);

    // h0 = relu(x @ W_in + b_in)
    gemm_wmma_f32<<<dim3(gRow, HID / 64), blk, 0, stream>>>(
        x, IN_DIM, W_in, HID, b_in, buf0, HID, N, IN_DIM, /*bias+relu*/3);

    // ---- GCN layer 1 ----
    gemm_wmma_f32<<<dim3(gRow, HID / 64), blk, 0, stream>>>(
        buf0, HID, W1, HID, nullptr, buf1, HID, N, HID, 0);
    agg_init    <<<gAgg,   blk, 0, stream>>>(buf1, dinv, b1, buf0, N);
    edge_scatter<<<gWaveE, blk, 0, stream>>>(src, dst, dinv, buf1, buf0, E);
    ln_relu     <<<gWaveN, blk, 0, stream>>>(buf0, g1, be1, N);

    // ---- GCN layer 2 ----
    gemm_wmma_f32<<<dim3(gRow, HID / 64), blk, 0, stream>>>(
        buf0, HID, W2, HID, nullptr, buf1, HID, N, HID, 0);
    agg_init    <<<gAgg,   blk, 0, stream>>>(buf1, dinv, b2, buf0, N);
    edge_scatter<<<gWaveE, blk, 0, stream>>>(src, dst, dinv, buf1, buf0, E);
    ln_relu     <<<gWaveN, blk, 0, stream>>>(buf0, g2, be2, N);

    // out = h2 @ W_out + b_out
    gemm_wmma_f32<<<dim3(gRow, OUT_DIM / 64), blk, 0, stream>>>(
        buf0, HID, W_out, OUT_DIM, b_out, (float*)d_out, OUT_DIM, N, HID, /*bias*/1);
}